// NWJ_26242250178717
// MI455X (gfx1250) — compile-verified
//
#include <hip/hip_runtime.h>
#include <hip/hip_bf16.h>

typedef __attribute__((ext_vector_type(2))) float v2f;
typedef __attribute__((ext_vector_type(8))) float v8f;

#define NROW 2048
#define DIM  768
#define KS   40
#define KP   48   // padded hidden size (3 x 16 WMMA N-tiles)

// ---------------------------------------------------------------------------
// Kernel 1: hx = X @ W1x.T  and  hyb = Y @ W1y.T + b1, via V_WMMA_F32_16X16X4_F32
// grid: (128 M-tiles, 3 N-tiles, 2 matrices), block: 1 wave (32 lanes)
// ---------------------------------------------------------------------------
__global__ __launch_bounds__(32) void nwj_gemm_wmma(
    const float* __restrict__ X, const float* __restrict__ Y,
    const float* __restrict__ W1, const float* __restrict__ b1,
    float* __restrict__ hxbuf, float* __restrict__ hybuf)
{
    const int lane = threadIdx.x;          // 0..31
    const int tm   = blockIdx.x;           // M tile
    const int tn   = blockIdx.y;           // N tile (0..2)
    const int mat  = blockIdx.z;           // 0 -> x path, 1 -> y path

    const float* A    = mat ? Y : X;
    const int    woff = mat ? DIM : 0;

    // A-matrix (16x4 f32) layout: lanes 0-15 hold M=lane, K={0,1};
    // lanes 16-31 hold M=lane-16, K={2,3}.
    const int mrow  = tm * 16 + (lane & 15);
    const int khalf = (lane >> 4) << 1;    // 0 or 2

    // B-matrix (4x16 f32): lane%16 = N column, same K-half split.
    const int ng  = tn * 16 + (lane & 15); // global hidden index
    const int ngc = (ng < KS) ? ng : (KS - 1);
    const bool nvalid = (ng < KS);

    const float* aptr = A  + (size_t)mrow * DIM + khalf;
    const float* bptr = W1 + (size_t)ngc * (2 * DIM) + woff + khalf;

    v8f c = {0.f, 0.f, 0.f, 0.f, 0.f, 0.f, 0.f, 0.f};

    #pragma unroll 4
    for (int kd = 0; kd < DIM; kd += 4) {
        v2f a, b;
        a.x = aptr[kd];
        a.y = aptr[kd + 1];
        float bx = bptr[kd];
        float by = bptr[kd + 1];
        b.x = nvalid ? bx : 0.0f;
        b.y = nvalid ? by : 0.0f;
        c = __builtin_amdgcn_wmma_f32_16x16x4_f32(
                /*neg_a=*/false, a, /*neg_b=*/false, b,
                /*c_mod=*/(short)0, c, /*reuse_a=*/false, /*reuse_b=*/false);
    }

    // C/D 16x16 f32 layout: VGPR r -> M=r (lanes 0-15) or M=8+r (lanes 16-31),
    // N = lane % 16.
    const int ncol  = tn * 16 + (lane & 15);
    const int mbase = tm * 16 + ((lane < 16) ? 0 : 8);
    #pragma unroll
    for (int r = 0; r < 8; ++r) {
        const int row = mbase + r;
        const float v = c[r];
        if (mat == 0) {
            hxbuf[row * KP + ncol] = v;
        } else {
            const float bias = (ncol < KS) ? b1[ncol] : 0.0f;
            hybuf[row * KP + ncol] = v + bias;
        }
    }
}

// ---------------------------------------------------------------------------
// Kernel 2: per 16x16 pair tile, acc = sum_k w2[k]*relu(hyb[i,k]+hx[j,k]),
// block-sum of exp(acc + b2 - 1) -> partial[block]
// ---------------------------------------------------------------------------
__global__ __launch_bounds__(256) void nwj_pair(
    const float* __restrict__ hxbuf, const float* __restrict__ hybuf,
    const float* __restrict__ W2, const float* __restrict__ b2p,
    float* __restrict__ partial)
{
    __shared__ float hyT[16][49];   // stride 49 -> conflict-free lane access
    __shared__ float hxT[16][49];
    __shared__ float w2s[KP];
    __shared__ float red[256];

    const int tid = threadIdx.x;
    const int bj  = blockIdx.x;     // x-sample tile (inner sum axis)
    const int bi  = blockIdx.y;     // y-sample tile

    for (int t = tid; t < 16 * KP; t += 256) {
        const int r = t / KP;
        const int k = t - r * KP;
        hyT[r][k] = hybuf[(bi * 16 + r) * KP + k];
        hxT[r][k] = hxbuf[(bj * 16 + r) * KP + k];
    }
    if (tid < KP) w2s[tid] = (tid < KS) ? W2[tid] : 0.0f;
    __syncthreads();

    const int tj = tid & 15;
    const int ti = tid >> 4;

    float acc = 0.0f;
    #pragma unroll
    for (int k = 0; k < KP; ++k) {
        const float s = hyT[ti][k] + hxT[tj][k];
        acc = fmaf(w2s[k], fmaxf(s, 0.0f), acc);
    }
    red[tid] = expf(acc + b2p[0] - 1.0f);
    __syncthreads();

    for (int s = 128; s > 0; s >>= 1) {
        if (tid < s) red[tid] += red[tid + s];
        __syncthreads();
    }
    if (tid == 0) partial[blockIdx.y * (NROW / 16) + blockIdx.x] = red[0];
}

// ---------------------------------------------------------------------------
// Kernel 3: diagonal (T0) term: sum_i sum_k w2[k]*relu(hyb[i,k]+hx[i,k])
// ---------------------------------------------------------------------------
__global__ __launch_bounds__(256) void nwj_diag(
    const float* __restrict__ hxbuf, const float* __restrict__ hybuf,
    const float* __restrict__ W2, float* __restrict__ t0partial)
{
    __shared__ float red[256];
    const int i = blockIdx.x * 256 + threadIdx.x;

    float acc = 0.0f;
    #pragma unroll
    for (int k = 0; k < KS; ++k) {
        const float s = hybuf[i * KP + k] + hxbuf[i * KP + k];
        acc = fmaf(W2[k], fmaxf(s, 0.0f), acc);
    }
    red[threadIdx.x] = acc;
    __syncthreads();
    for (int s = 128; s > 0; s >>= 1) {
        if (threadIdx.x < s) red[threadIdx.x] += red[threadIdx.x + s];
        __syncthreads();
    }
    if (threadIdx.x == 0) t0partial[blockIdx.x] = red[0];
}

// ---------------------------------------------------------------------------
// Kernel 4: deterministic final reduction + scalar combine
// lower = (sumT0/N + b2) - sumExp/N^2     (exp term already includes b2-1)
// ---------------------------------------------------------------------------
__global__ __launch_bounds__(256) void nwj_final(
    const float* __restrict__ partial, const float* __restrict__ t0partial,
    const float* __restrict__ b2p, float* __restrict__ out)
{
    __shared__ float red[256];
    const int nPart = (NROW / 16) * (NROW / 16);   // 16384

    float s = 0.0f;
    for (int t = threadIdx.x; t < nPart; t += 256) s += partial[t];
    red[threadIdx.x] = s;
    __syncthreads();
    for (int r = 128; r > 0; r >>= 1) {
        if (threadIdx.x < r) red[threadIdx.x] += red[threadIdx.x + r];
        __syncthreads();
    }
    if (threadIdx.x == 0) {
        const float sumExp = red[0];
        float sumT0 = 0.0f;
        for (int t = 0; t < NROW / 256; ++t) sumT0 += t0partial[t];
        const float Nf = (float)NROW;
        out[0] = (sumT0 / Nf + b2p[0]) - sumExp / (Nf * Nf);
    }
}

// ---------------------------------------------------------------------------
extern "C" void kernel_launch(void* const* d_in, const int* in_sizes, int n_in,
                              void* d_out, int out_size, void* d_ws, size_t ws_size,
                              hipStream_t stream) {
    const float* X  = (const float*)d_in[0];   // (2048, 768)
    const float* Y  = (const float*)d_in[1];   // (2048, 768)
    const float* W1 = (const float*)d_in[2];   // (40, 1536)
    const float* b1 = (const float*)d_in[3];   // (40,)
    const float* W2 = (const float*)d_in[4];   // (1, 40)
    const float* b2 = (const float*)d_in[5];   // (1,)
    float* out = (float*)d_out;

    float* ws      = (float*)d_ws;
    float* hxbuf   = ws;                       // 2048*48
    float* hybuf   = hxbuf + NROW * KP;        // 2048*48
    float* partial = hybuf + NROW * KP;        // 16384
    float* t0part  = partial + (NROW / 16) * (NROW / 16);   // 8

    nwj_gemm_wmma<<<dim3(NROW / 16, KP / 16, 2), 32, 0, stream>>>(
        X, Y, W1, b1, hxbuf, hybuf);

    nwj_pair<<<dim3(NROW / 16, NROW / 16), 256, 0, stream>>>(
        hxbuf, hybuf, W2, b2, partial);

    nwj_diag<<<dim3(NROW / 256), 256, 0, stream>>>(hxbuf, hybuf, W2, t0part);

    nwj_final<<<1, 256, 0, stream>>>(partial, t0part, b2, out);
}